// PoseGraph_50337016709659
// MI455X (gfx1250) — compile-verified
//
#include <hip/hip_runtime.h>
#include <math.h>

#define TPB 256

typedef unsigned int u32x4 __attribute__((ext_vector_type(4)));
typedef int          i32x8 __attribute__((ext_vector_type(8)));
typedef int          i32x4 __attribute__((ext_vector_type(4)));

// ---------------- SE(3) math helpers (mirror the JAX reference) ----------------

__device__ __forceinline__ float3 qrot(float4 q, float3 v) {
    // uv = cross(q.xyz, v); uuv = cross(q.xyz, uv); v + 2*(w*uv + uuv)
    float uvx = q.y * v.z - q.z * v.y;
    float uvy = q.z * v.x - q.x * v.z;
    float uvz = q.x * v.y - q.y * v.x;
    float uuvx = q.y * uvz - q.z * uvy;
    float uuvy = q.z * uvx - q.x * uvz;
    float uuvz = q.x * uvy - q.y * uvx;
    return make_float3(v.x + 2.0f * (q.w * uvx + uuvx),
                       v.y + 2.0f * (q.w * uvy + uuvy),
                       v.z + 2.0f * (q.w * uvz + uuvz));
}

__device__ __forceinline__ float4 qmul(float4 a, float4 b) {
    return make_float4(
        a.w * b.x + a.x * b.w + a.y * b.z - a.z * b.y,
        a.w * b.y - a.x * b.z + a.y * b.w + a.z * b.x,
        a.w * b.z + a.x * b.y - a.y * b.x + a.z * b.w,
        a.w * b.w - a.x * b.x - a.y * b.y - a.z * b.z);
}

// se3_log of pose (t, q) -> writes 6 floats [tau, phi] to o
__device__ __forceinline__ void se3_log(float3 t, float4 q, float* o) {
    if (q.w < 0.0f) { q.x = -q.x; q.y = -q.y; q.z = -q.z; q.w = -q.w; }
    float n2 = q.x * q.x + q.y * q.y + q.z * q.z;
    float n  = sqrtf(n2);
    float safe_n = fmaxf(n, 1e-9f);
    float factor = (n < 1e-6f) ? (2.0f / fmaxf(q.w, 1e-9f))
                               : (2.0f * atan2f(n, q.w) / safe_n);
    float px = factor * q.x, py = factor * q.y, pz = factor * q.z;
    float theta2 = px * px + py * py + pz * pz;
    float theta  = sqrtf(fmaxf(theta2, 1e-24f));
    float half   = 0.5f * theta;
    float coef;
    if (theta < 1e-4f) {
        coef = 1.0f / 12.0f;
    } else {
        float s = sinf(half);
        float c = cosf(half);
        coef = (1.0f - half * c / fmaxf(s, 1e-9f)) / fmaxf(theta2, 1e-24f);
    }
    // pt = cross(phi, t)
    float ptx = py * t.z - pz * t.y;
    float pty = pz * t.x - px * t.z;
    float ptz = px * t.y - py * t.x;
    // pp = cross(phi, pt)
    float ppx = py * ptz - pz * pty;
    float ppy = pz * ptx - px * ptz;
    float ppz = px * pty - py * ptx;
    o[0] = t.x - 0.5f * ptx + coef * ppx;
    o[1] = t.y - 0.5f * pty + coef * ppy;
    o[2] = t.z - 0.5f * ptz + coef * ppz;
    o[3] = px; o[4] = py; o[5] = pz;
}

// ---------------- main kernel ----------------

__global__ __launch_bounds__(TPB) void pose_graph_kernel(
    const float* __restrict__ nodes,   // [N_NODES, 7]
    const int*   __restrict__ edges,   // [E, 2]
    const float* __restrict__ rel,     // [E, 7]
    float*       __restrict__ out,     // [6 + 6*E]
    int E)
{
    __shared__ float s_rp[TPB * 7];    // staged relative-pose tile (TDM target)
    __shared__ float s_out[TPB * 6];   // staged output tile (coalesced NT store)

    const int tid = threadIdx.x;
    const int e0  = blockIdx.x * TPB;
    const int n   = min(TPB, E - e0);
    const int nf  = n * 7;             // floats in this block's rel-pose tile

    // --- TDM: async DMA of the contiguous rel-pose tile into LDS (wave 0 issues) ---
    if (tid == 0) {
        unsigned lds_off = (unsigned)(unsigned long long)(&s_rp[0]);
        unsigned long long ga =
            (unsigned long long)(size_t)(rel + (size_t)e0 * 7);

        u32x4 g0;
        g0[0] = 1u;                                         // count=1 (valid user D#)
        g0[1] = lds_off;                                    // lds_addr (bytes)
        g0[2] = (unsigned)(ga & 0xFFFFFFFFull);             // global_addr[31:0]
        g0[3] = (unsigned)((ga >> 32) & 0x01FFFFFFull)      // global_addr[56:32]
              | (2u << 30);                                 // type=2 ("image")

        i32x8 g1;
        g1[0] = (int)(2u << 16);                            // data_size=4B; wg_mask=0
        g1[1] = (int)(((unsigned)nf & 0xFFFFu) << 16);      // tensor_dim0[15:0]
        g1[2] = (int)((((unsigned)nf >> 16) & 0xFFFFu)      // tensor_dim0[31:16]
              | (1u << 16));                                // tensor_dim1 = 1 (lo)
        g1[3] = (int)(((unsigned)nf & 0xFFFFu) << 16);      // tile_dim0 = nf
        g1[4] = 1;                                          // tile_dim1=1, tile_dim2=0
        g1[5] = nf;                                         // tensor_dim0_stride (lo)
        g1[6] = 0;
        g1[7] = 0;

        i32x4 z4 = {0, 0, 0, 0};                            // groups 2/3 unused (<=2D)
        i32x8 z8 = {0, 0, 0, 0, 0, 0, 0, 0};                // 6-arg form (clang-23)
        __builtin_amdgcn_tensor_load_to_lds(g0, g1, z4, z4, z8, 0);
    }

    // --- overlap: per-thread edge + node gathers (node table is L2-resident) ---
    const bool active = tid < n;
    int2 ed = make_int2(0, 0);
    if (active) ed = ((const int2*)edges)[e0 + tid];        // global_load_b64, coalesced

    const float* n1p = nodes + (size_t)ed.x * 7;
    const float* n2p = nodes + (size_t)ed.y * 7;
    float3 t1 = make_float3(n1p[0], n1p[1], n1p[2]);
    float4 q1 = make_float4(n1p[3], n1p[4], n1p[5], n1p[6]);
    float3 t2 = make_float3(n2p[0], n2p[1], n2p[2]);
    float4 q2 = make_float4(n2p[3], n2p[4], n2p[5], n2p[6]);

    // --- wait for TDM, then make tile visible to the whole workgroup ---
    if (tid == 0) __builtin_amdgcn_s_wait_tensorcnt(0);
    __syncthreads();

    if (active) {
        const float* rp = &s_rp[tid * 7];                   // stride 7 -> conflict-free
        float3 rt = make_float3(rp[0], rp[1], rp[2]);
        float4 rq = make_float4(rp[3], rp[4], rp[5], rp[6]);

        // inv(node1): qi = conj(q1); ti = -qrot(qi, t1)
        float4 qi = make_float4(-q1.x, -q1.y, -q1.z, q1.w);
        float3 ti = qrot(qi, t1);
        ti.x = -ti.x; ti.y = -ti.y; ti.z = -ti.z;

        // A = rel * inv(node1)
        float3 ra = qrot(rq, ti);
        float3 tA = make_float3(rt.x + ra.x, rt.y + ra.y, rt.z + ra.z);
        float4 qA = qmul(rq, qi);

        // err = A * node2
        float3 rb = qrot(qA, t2);
        float3 tE = make_float3(tA.x + rb.x, tA.y + rb.y, tA.z + rb.z);
        float4 qE = qmul(qA, q2);

        se3_log(tE, qE, &s_out[tid * 6]);
    }
    __syncthreads();

    // --- coalesced non-temporal stores of the staged output tile ---
    {
        float* ob = out + 6 + (size_t)e0 * 6;
        const int total = n * 6;
        for (int idx = tid; idx < total; idx += TPB)
            __builtin_nontemporal_store(s_out[idx], &ob[idx]);
    }

    // --- prior error: se3_log(nodes[0]) -> out[0..5] ---
    if (blockIdx.x == 0 && tid == 0) {
        float o[6];
        se3_log(make_float3(nodes[0], nodes[1], nodes[2]),
                make_float4(nodes[3], nodes[4], nodes[5], nodes[6]), o);
        #pragma unroll
        for (int k = 0; k < 6; ++k) out[k] = o[k];
    }
}

// ---------------- launcher ----------------

extern "C" void kernel_launch(void* const* d_in, const int* in_sizes, int n_in,
                              void* d_out, int out_size, void* d_ws, size_t ws_size,
                              hipStream_t stream) {
    const float* nodes = (const float*)d_in[0];
    const int*   edges = (const int*)d_in[1];
    const float* rel   = (const float*)d_in[2];
    float*       out   = (float*)d_out;

    const int E = in_sizes[1] / 2;
    const int grid = (E + TPB - 1) / TPB;

    hipLaunchKernelGGL(pose_graph_kernel, dim3(grid), dim3(TPB), 0, stream,
                       nodes, edges, rel, out, E);
}